// FoldingTrunk_592705487204
// MI455X (gfx1250) — compile-verified
//
#include <hip/hip_runtime.h>
#include <hip/hip_bf16.h>
#include <math.h>

// ---------------- problem constants ----------------
#define B_    2
#define LSEQ  768
#define S_IN  1280
#define S_OUT 32
#define DINNER 64
#define DSTATE 16
#define DCONV  4
#define DTRANK 2
#define M_TOT (B_ * LSEQ)          // 1536 rows
#define NCHUNK 16
#define CLEN   (LSEQ / NCHUNK)     // 48

typedef float v2f __attribute__((ext_vector_type(2)));
typedef float v4f __attribute__((ext_vector_type(4)));
typedef float v8f __attribute__((ext_vector_type(8)));

// =====================================================================
// f32 GEMM via V_WMMA_F32_16X16X4_F32:  C[M,N] = A[M,K] * W[N,K]^T (+bias)
// One wave computes one 16x16 tile. EXEC all-ones (32 uniform threads).
// A 16x4 layout: lanes 0-15 = rows, lane-half selects K pair; B 4x16 dual.
// =====================================================================
__global__ void gemm_wmma_f32(const float* __restrict__ A,
                              const float* __restrict__ W,
                              const float* __restrict__ bias,
                              float* __restrict__ C,
                              int M, int N, int K) {
  const int m0   = blockIdx.x * 16;
  const int n0   = blockIdx.y * 16;
  const int lane = threadIdx.x;      // 0..31
  const int half = lane >> 4;        // 0 or 1
  const int lr   = lane & 15;

  v8f acc = {};
  const float* arow = A + (m0 + lr) * K + half * 2;
  const float* brow = W + (n0 + lr) * K + half * 2;

#pragma unroll 4
  for (int k0 = 0; k0 < K; k0 += 4) {
    v2f a = *(const v2f*)(arow + k0);   // A[m0+lr, k0+2*half .. +1]
    v2f b = *(const v2f*)(brow + k0);   // B[k0+2*half.., n0+lr] = W[n0+lr, ...]
    acc = __builtin_amdgcn_wmma_f32_16x16x4_f32(
        /*neg_a=*/false, a, /*neg_b=*/false, b,
        /*c_mod=*/(short)0, acc, /*reuse_a=*/false, /*reuse_b=*/false);
  }

  const float bv = bias ? bias[n0 + lr] : 0.0f;
#pragma unroll
  for (int r = 0; r < 8; ++r) {
    C[(m0 + r + 8 * half) * N + n0 + lr] = acc[r] + bv;
  }
}

// =====================================================================
// in_proj: xz = u @ in_proj_w^T ; split into x_pre (0..63) / gate (64..127)
// =====================================================================
__global__ void in_proj_kernel(const float* __restrict__ u,
                               const float* __restrict__ w,
                               float* __restrict__ x_pre,
                               float* __restrict__ gate) {
  const int m = blockIdx.x;          // row 0..1535
  const int c = threadIdx.x;         // 0..127
  const float* ur = u + m * S_OUT;
  const float* wr = w + c * S_OUT;
  float acc = 0.f;
#pragma unroll
  for (int j = 0; j < S_OUT; ++j) acc = fmaf(ur[j], wr[j], acc);
  if (c < DINNER) x_pre[m * DINNER + c] = acc;
  else            gate[m * DINNER + (c - DINNER)] = acc;
}

// =====================================================================
// causal depthwise conv + SiLU, then x_proj -> (dt_in, B, C), then
// dt = softplus(dt_in @ dt_proj_w^T + dt_proj_b).  One block per row m.
// =====================================================================
__global__ void conv_proj_kernel(const float* __restrict__ x_pre,
                                 const float* __restrict__ conv_w,
                                 const float* __restrict__ conv_b,
                                 const float* __restrict__ x_proj_w,
                                 const float* __restrict__ dt_w,
                                 const float* __restrict__ dt_b,
                                 float* __restrict__ xo,
                                 float* __restrict__ dto,
                                 float* __restrict__ Bo,
                                 float* __restrict__ Co) {
  const int m = blockIdx.x;              // b*768 + l
  const int l = m % LSEQ;
  const int t = threadIdx.x;             // 0..63 (channel d)
  __shared__ float sx[DINNER];
  __shared__ float sdbl[DTRANK];

  float acc = conv_b[t];
#pragma unroll
  for (int k = 0; k < DCONV; ++k) {
    const int ls = l - (DCONV - 1) + k;
    if (ls >= 0) acc = fmaf(x_pre[(m - (DCONV - 1) + k) * DINNER + t],
                            conv_w[t * DCONV + k], acc);
  }
  const float xv = acc / (1.0f + __expf(-acc));   // SiLU
  sx[t] = xv;
  xo[m * DINNER + t] = xv;
  __syncthreads();

  if (t < DTRANK + 2 * DSTATE) {                  // 34 rows of x_proj_w
    const float* pr = x_proj_w + t * DINNER;
    float a = 0.f;
#pragma unroll
    for (int d = 0; d < DINNER; ++d) a = fmaf(sx[d], pr[d], a);
    if (t < DTRANK)               sdbl[t] = a;
    else if (t < DTRANK + DSTATE) Bo[m * DSTATE + (t - DTRANK)] = a;
    else                          Co[m * DSTATE + (t - DTRANK - DSTATE)] = a;
  }
  __syncthreads();

  float dr = fmaf(sdbl[0], dt_w[t * DTRANK + 0],
             fmaf(sdbl[1], dt_w[t * DTRANK + 1], dt_b[t]));
  float dtv = (dr > 20.0f) ? dr : log1pf(__expf(dr));   // softplus
  dto[m * DINNER + t] = dtv;
}

// =====================================================================
// Scan phase 1: per (b,d,n,chunk) compute P = prod(a_l), Q = local scan tail
// h_out = P*h_in + Q   with a_l = exp(dt*A), u_l = dt*B*x
// =====================================================================
__global__ void scan_part1(const float* __restrict__ dt,
                           const float* __restrict__ Bm,
                           const float* __restrict__ x,
                           const float* __restrict__ A_log,
                           float* __restrict__ P, float* __restrict__ Q) {
  const int g = blockIdx.x * blockDim.x + threadIdx.x;   // 32768
  const int n = g & 15;
  const int c = (g >> 4) & 15;
  const int d = (g >> 8) & 63;
  const int b = g >> 14;
  const float Adn = -__expf(A_log[d * DSTATE + n]);
  float p = 1.0f, q = 0.0f;
  const int l0 = c * CLEN;
  for (int j = 0; j < CLEN; ++j) {
    const int m = b * LSEQ + l0 + j;
    const float dtv = dt[m * DINNER + d];
    const float a = __expf(dtv * Adn);
    const float uu = dtv * Bm[m * DSTATE + n] * x[m * DINNER + d];
    p *= a;
    q = fmaf(a, q, uu);
  }
  const int flat = (b * DINNER + d) * DSTATE + n;
  P[flat * NCHUNK + c] = p;
  Q[flat * NCHUNK + c] = q;
}

// Scan phase 2: combine chunk boundaries -> start state per chunk
__global__ void scan_part2(const float* __restrict__ P,
                           const float* __restrict__ Q,
                           float* __restrict__ hs) {
  const int flat = blockIdx.x * blockDim.x + threadIdx.x;   // 2048
  float h = 0.f;
#pragma unroll
  for (int c = 0; c < NCHUNK; ++c) {
    hs[flat * NCHUNK + c] = h;
    h = fmaf(P[flat * NCHUNK + c], h, Q[flat * NCHUNK + c]);
  }
}

// Scan phase 3: replay chunk with known start state; reduce over n (16 lanes)
// y = sum_n h*C + x*D, then gate with SiLU(gate)
__global__ void scan_part3(const float* __restrict__ dt,
                           const float* __restrict__ Bm,
                           const float* __restrict__ Cm,
                           const float* __restrict__ x,
                           const float* __restrict__ gate,
                           const float* __restrict__ A_log,
                           const float* __restrict__ Dvec,
                           const float* __restrict__ hs,
                           float* __restrict__ yg) {
  const int g = blockIdx.x * blockDim.x + threadIdx.x;   // 32768
  const int n = g & 15;
  const int c = (g >> 4) & 15;
  const int d = (g >> 8) & 63;
  const int b = g >> 14;
  const float Adn = -__expf(A_log[d * DSTATE + n]);
  const int flat = (b * DINNER + d) * DSTATE + n;
  float h = hs[flat * NCHUNK + c];
  const float Dd = Dvec[d];
  const int l0 = c * CLEN;
  for (int j = 0; j < CLEN; ++j) {
    const int m = b * LSEQ + l0 + j;
    const float dtv = dt[m * DINNER + d];
    const float a = __expf(dtv * Adn);
    const float xv = x[m * DINNER + d];
    h = fmaf(a, h, dtv * Bm[m * DSTATE + n] * xv);
    float contrib = h * Cm[m * DSTATE + n];
    // reduce across the 16 state lanes (wave32, width-16 segments)
    contrib += __shfl_xor(contrib, 1, 16);
    contrib += __shfl_xor(contrib, 2, 16);
    contrib += __shfl_xor(contrib, 4, 16);
    contrib += __shfl_xor(contrib, 8, 16);
    if (n == 0) {
      const float y = contrib + xv * Dd;
      const float gv = gate[m * DINNER + d];
      const float sg = gv / (1.0f + __expf(-gv));   // SiLU(gate)
      yg[m * DINNER + d] = y * sg;
    }
  }
}

// =====================================================================
// z kernel (bandwidth dominant, ~151 MB out): NT 128-bit streaming stores.
// z_prime[b,s,t,f] = z[b,s,t]*zw[f] + zb[f] + sp[b,s,f]*sp[b,t,f]
// =====================================================================
__global__ void z_outer_kernel(const float* __restrict__ z,
                               const float* __restrict__ zw,
                               const float* __restrict__ zb,
                               const float* __restrict__ sp,
                               float* __restrict__ out) {
  const int i = blockIdx.x * blockDim.x + threadIdx.x;  // 9,437,184 threads
  const int f4 = i & 7;                 // which float4 of the 32 features
  const int rest = i >> 3;              // (b*768+s)*768 + t
  const int t  = rest % LSEQ;
  const int bs = rest / LSEQ;           // b*768 + s
  const int b  = bs / LSEQ;

  const float zv = z[rest];
  const v4f w  = ((const v4f*)zw)[f4];
  const v4f bi = ((const v4f*)zb)[f4];
  const v4f aa = ((const v4f*)sp)[bs * 8 + f4];            // sp[b,s,f4]
  const v4f cc = ((const v4f*)sp)[(b * LSEQ + t) * 8 + f4]; // sp[b,t,f4]
  const v4f r = zv * w + bi + aa * cc;
  __builtin_nontemporal_store(r, ((v4f*)out) + i);
}

// =====================================================================
extern "C" void kernel_launch(void* const* d_in, const int* in_sizes, int n_in,
                              void* d_out, int out_size, void* d_ws, size_t ws_size,
                              hipStream_t stream) {
  const float* s         = (const float*)d_in[0];
  const float* z         = (const float*)d_in[1];
  const float* s_proj_w  = (const float*)d_in[2];
  const float* s_proj_b  = (const float*)d_in[3];
  const float* z_proj_w  = (const float*)d_in[4];
  const float* z_proj_b  = (const float*)d_in[5];
  const float* in_proj_w = (const float*)d_in[6];
  const float* conv_w    = (const float*)d_in[7];
  const float* conv_b    = (const float*)d_in[8];
  const float* x_proj_w  = (const float*)d_in[9];
  const float* dt_proj_w = (const float*)d_in[10];
  const float* dt_proj_b = (const float*)d_in[11];
  const float* A_log     = (const float*)d_in[12];
  const float* Dvec      = (const float*)d_in[13];
  const float* out_proj_w= (const float*)d_in[14];

  float* out = (float*)d_out;
  float* s_prime = out;                              // [2,768,32]
  float* z_prime = out + M_TOT * S_OUT;              // [2,768,768,32]

  // workspace layout (floats)
  float* ws    = (float*)d_ws;
  float* u     = ws;                     // 49152   : s_proj output
  float* x_pre = u     + M_TOT * S_OUT;  // 98304
  float* gate  = x_pre + M_TOT * DINNER; // 98304
  float* xconv = gate  + M_TOT * DINNER; // 98304
  float* dt    = xconv + M_TOT * DINNER; // 98304
  float* Bm    = dt    + M_TOT * DINNER; // 24576
  float* Cm    = Bm    + M_TOT * DSTATE; // 24576
  float* P     = Cm    + M_TOT * DSTATE; // 32768
  float* Q     = P     + B_ * DINNER * DSTATE * NCHUNK; // 32768
  float* hs    = Q     + B_ * DINNER * DSTATE * NCHUNK; // 32768
  float* yg    = hs    + B_ * DINNER * DSTATE * NCHUNK; // 98304

  // 1) s_proj: WMMA f32 GEMM  [1536,1280] x [32,1280]^T (+bias)
  gemm_wmma_f32<<<dim3(M_TOT / 16, S_OUT / 16), 32, 0, stream>>>(
      s, s_proj_w, s_proj_b, u, M_TOT, S_OUT, S_IN);

  // 2) in_proj -> x_pre, gate
  in_proj_kernel<<<M_TOT, 2 * DINNER, 0, stream>>>(u, in_proj_w, x_pre, gate);

  // 3) conv + SiLU + x_proj + dt
  conv_proj_kernel<<<M_TOT, DINNER, 0, stream>>>(
      x_pre, conv_w, conv_b, x_proj_w, dt_proj_w, dt_proj_b, xconv, dt, Bm, Cm);

  // 4) chunked parallel scan
  scan_part1<<<(B_ * DINNER * DSTATE * NCHUNK) / 256, 256, 0, stream>>>(
      dt, Bm, xconv, A_log, P, Q);
  scan_part2<<<(B_ * DINNER * DSTATE) / 256, 256, 0, stream>>>(P, Q, hs);
  scan_part3<<<(B_ * DINNER * DSTATE * NCHUNK) / 256, 256, 0, stream>>>(
      dt, Bm, Cm, xconv, gate, A_log, Dvec, hs, yg);

  // 5) out_proj: WMMA f32 GEMM  [1536,64] x [32,64]^T -> s_prime (d_out)
  gemm_wmma_f32<<<dim3(M_TOT / 16, S_OUT / 16), 32, 0, stream>>>(
      yg, out_proj_w, nullptr, s_prime, M_TOT, S_OUT, DINNER);

  // 6) z projection + outer product, NT streaming stores (151 MB)
  const int zthreads = B_ * LSEQ * LSEQ * (S_OUT / 4);   // 9,437,184
  z_outer_kernel<<<zthreads / 256, 256, 0, stream>>>(
      z, z_proj_w, z_proj_b, s_prime, z_prime);
}